// simclr_20804821582052
// MI455X (gfx1250) — compile-verified
//
#include <hip/hip_runtime.h>

typedef __attribute__((ext_vector_type(2))) float v2f;
typedef __attribute__((ext_vector_type(8))) float v8f;

// ---------------------------------------------------------------------------
// zero a float buffer
__global__ void k_zero(float* __restrict__ p, int n) {
    int i = blockIdx.x * blockDim.x + threadIdx.x;
    if (i < n) p[i] = 0.0f;
}

// ---------------------------------------------------------------------------
// z = (1 + eps[layer]) * h   (float4 vectorized; n4 = N*64/4)
__global__ void k_scale_init(const float* __restrict__ h, float* __restrict__ z,
                             const float* __restrict__ eps_arr, int layer, int n4) {
    int i = blockIdx.x * blockDim.x + threadIdx.x;
    if (i >= n4) return;
    float s = 1.0f + eps_arr[layer];
    float4 v = ((const float4*)h)[i];
    v.x *= s; v.y *= s; v.z *= s; v.w *= s;
    ((float4*)z)[i] = v;
}

// ---------------------------------------------------------------------------
// z[row[e], :] += h[col[e], :]   (thread = (edge, 4-feature chunk))
__global__ void k_scatter(const int* __restrict__ rows, const int* __restrict__ cols,
                          const float* __restrict__ h, float* __restrict__ z, int E) {
    long long tid = (long long)blockIdx.x * blockDim.x + threadIdx.x;
    if (tid >= (long long)E * 16) return;
    int e  = (int)(tid >> 4);
    int fc = ((int)tid & 15) << 2;
    int src = cols[e];
    int dst = rows[e];
    const float4 v = *(const float4*)(h + (size_t)src * 64 + fc);
    float* zp = z + (size_t)dst * 64 + fc;
    atomicAdd(zp + 0, v.x);
    atomicAdd(zp + 1, v.y);
    atomicAdd(zp + 2, v.z);
    atomicAdd(zp + 3, v.w);
}

// ---------------------------------------------------------------------------
// out[M x Nc] = epilogue( A[M x K] @ W[K x Nc] + bias )
// flags bit0: relu after bias;  bit1: then add bias2 and relu again
// One wave computes a 16x64 D strip (4 col tiles) so each A fragment load
// feeds 4 V_WMMA_F32_16X16X4_F32 issues; A is read once per 64-column strip.
// Fragment layouts per CDNA5 ISA 7.12.2:
//   A 16x4 : lanes 0-15 rows M, VGPR{0,1} = K{0,1}; lanes 16-31 = K{2,3}
//   B 4x16 : lanes 0-15 cols N, VGPR{0,1} = K{0,1}; lanes 16-31 = K{2,3}
//   C 16x16: VGPR j -> row j (lanes 0-15) / row j+8 (lanes 16-31), col = lane%16
__global__ void k_gemm_wmma(const float* __restrict__ A, const float* __restrict__ W,
                            const float* __restrict__ bias, const float* __restrict__ bias2,
                            float* __restrict__ out, int M, int K, int Nc, int flags) {
    const int wave = threadIdx.x >> 5;     // 0..7 (wave32)
    const int lane = threadIdx.x & 31;
    const int half = lane >> 4;            // 0 or 1
    const int lm   = lane & 15;
    const int rowTile = blockIdx.x * 8 + wave;
    if (rowTile * 16 >= M) return;         // wave-uniform: EXEC stays all-ones
    const int r0 = rowTile * 16;
    const int c0 = blockIdx.y * 64;        // 4 x 16-wide col tiles per wave

    v8f acc[4];
#pragma unroll
    for (int t = 0; t < 4; ++t) acc[t] = (v8f){0.f,0.f,0.f,0.f,0.f,0.f,0.f,0.f};

    const float* __restrict__ aptr = A + (size_t)(r0 + lm) * K + 2 * half;
    const float* __restrict__ wptr = W + (size_t)(2 * half) * Nc + c0 + lm;

    for (int k = 0; k < K; k += 4) {
        v2f a;
        { float2 t2 = *(const float2*)(aptr + k); a.x = t2.x; a.y = t2.y; }
        const float* __restrict__ w0 = wptr + (size_t)k * Nc;
        const float* __restrict__ w1 = wptr + (size_t)(k + 1) * Nc;
#pragma unroll
        for (int t = 0; t < 4; ++t) {
            v2f b;
            b.x = w0[t * 16];
            b.y = w1[t * 16];
            acc[t] = __builtin_amdgcn_wmma_f32_16x16x4_f32(false, a, false, b,
                                                           (short)0, acc[t],
                                                           false, false);
        }
    }

#pragma unroll
    for (int t = 0; t < 4; ++t) {
        const int cc = c0 + t * 16 + lm;
        const float bv  = bias  ? bias[cc]  : 0.0f;
        const float b2v = bias2 ? bias2[cc] : 0.0f;
        float* op = out + (size_t)(r0 + 8 * half) * Nc + cc;
#pragma unroll
        for (int j = 0; j < 8; ++j) {
            float v = acc[t][j] + bv;
            if (flags & 1) v = fmaxf(v, 0.0f);
            if (flags & 2) v = fmaxf(v + b2v, 0.0f);
            op[(size_t)j * Nc] = v;
        }
    }
}

// ---------------------------------------------------------------------------
// per-feature sum / sum-of-squares over N rows (64 features)
__global__ void k_bn_stats(const float* __restrict__ z, float* __restrict__ stats, int Nn) {
    __shared__ float ssum[256];
    __shared__ float ssq[256];
    const int f  = threadIdx.x & 63;
    const int rg = threadIdx.x >> 6;   // 0..3 row groups
    float s = 0.0f, q = 0.0f;
    for (int n = blockIdx.x * 4 + rg; n < Nn; n += gridDim.x * 4) {
        float v = z[(size_t)n * 64 + f];
        s += v;
        q += v * v;
    }
    ssum[threadIdx.x] = s;
    ssq[threadIdx.x]  = q;
    __syncthreads();
    if (threadIdx.x < 64) {
        s = ssum[f] + ssum[f + 64] + ssum[f + 128] + ssum[f + 192];
        q = ssq[f]  + ssq[f + 64]  + ssq[f + 128]  + ssq[f + 192];
        atomicAdd(&stats[f], s);
        atomicAdd(&stats[64 + f], q);
    }
}

// stats[0..63]=sum, stats[64..127]=sumsq  ->  ss[0..63]=scale, ss[64..127]=shift
__global__ void k_bn_finalize(const float* __restrict__ stats, const float* __restrict__ gamma,
                              const float* __restrict__ beta, float* __restrict__ ss, float invN) {
    int f = threadIdx.x;               // 64 threads
    float mean  = stats[f] * invN;
    float var   = stats[64 + f] * invN - mean * mean;
    float scale = gamma[f] * rsqrtf(var + 1e-5f);
    ss[f]      = scale;
    ss[64 + f] = beta[f] - mean * scale;
}

// h = z*scale + shift ; pooled[batch[n], layer*64 + f] += h
__global__ void k_bn_apply_pool(const float* __restrict__ z, const float* __restrict__ ss,
                                float* __restrict__ h, const int* __restrict__ batch,
                                float* __restrict__ pooled, int layer, int Nn) {
    int tid = blockIdx.x * blockDim.x + threadIdx.x;
    if (tid >= Nn * 64) return;
    int n = tid >> 6, f = tid & 63;
    float v = z[tid] * ss[f] + ss[64 + f];
    h[tid] = v;
    atomicAdd(&pooled[(size_t)batch[n] * 192 + layer * 64 + f], v);
}

// ---------------------------------------------------------------------------
static inline int cdiv(long long a, long long b) { return (int)((a + b - 1) / b); }

extern "C" void kernel_launch(void* const* d_in, const int* in_sizes, int n_in,
                              void* d_out, int out_size, void* d_ws, size_t ws_size,
                              hipStream_t stream) {
    const float* x       = (const float*)d_in[0];
    const int*   eidx    = (const int*)d_in[1];
    const int*   batch   = (const int*)d_in[2];
    const float* W1      = (const float*)d_in[3];
    const float* b1      = (const float*)d_in[4];
    const float* W2      = (const float*)d_in[5];
    const float* b2      = (const float*)d_in[6];
    const float* gin_eps = (const float*)d_in[7];
    const float* gin_b   = (const float*)d_in[8];
    const float* bn_g    = (const float*)d_in[9];
    const float* bn_b    = (const float*)d_in[10];
    const float* Wp1     = (const float*)d_in[11];
    const float* bp1     = (const float*)d_in[12];
    const float* Wp2     = (const float*)d_in[13];
    const float* bp2     = (const float*)d_in[14];
    float*       out     = (float*)d_out;

    const int D  = 64;
    const int L  = 3;
    const int LD = L * D;                 // 192
    const int N  = in_sizes[0] / D;       // 100000
    const int E  = in_sizes[1] / 2;       // 800000
    const int NG = out_size / LD;         // 512

    const int* rows = eidx;
    const int* cols = eidx + E;

    // workspace layout (floats)
    float* bufZ   = (float*)d_ws;                  // N*64
    float* bufH   = bufZ + (size_t)N * D;          // N*64
    float* pooled = bufH + (size_t)N * D;          // NG*192
    float* tproj  = pooled + (size_t)NG * LD;      // NG*192
    float* stats  = tproj + (size_t)NG * LD;       // 256 (sum/sumsq then scale/shift)

    // zero pooled accumulator
    k_zero<<<cdiv((long long)NG * LD, 256), 256, 0, stream>>>(pooled, NG * LD);

    const dim3 gemmGridN(cdiv(N, 128), D / 64);    // y = 1 (64-wide strip per wave)
    const dim3 gemmGridP(cdiv(NG, 128), LD / 64);  // y = 3

    const float* h_in = x;
    for (int i = 0; i < L; ++i) {
        // z = (1+eps)*h + scatter-sum over edges
        k_scale_init<<<cdiv((long long)N * 16, 256), 256, 0, stream>>>(
            h_in, bufZ, gin_eps, i, N * 16);
        k_scatter<<<cdiv((long long)E * 16, 256), 256, 0, stream>>>(
            rows, cols, h_in, bufZ, E);
        // z1 = relu(z @ W1 + b1)
        k_gemm_wmma<<<gemmGridN, 256, 0, stream>>>(
            bufZ, W1 + (size_t)i * D * D, b1 + (size_t)i * D, nullptr,
            bufH, N, D, D, 1);
        // z2 = relu(relu(z1 @ W2 + b2) + gin_bias)
        k_gemm_wmma<<<gemmGridN, 256, 0, stream>>>(
            bufH, W2 + (size_t)i * D * D, b2 + (size_t)i * D, gin_b + (size_t)i * D,
            bufZ, N, D, D, 3);
        // batch-norm statistics, finalize, apply + pooled segment-sum
        k_zero<<<1, 128, 0, stream>>>(stats, 128);
        k_bn_stats<<<512, 256, 0, stream>>>(bufZ, stats, N);
        k_bn_finalize<<<1, 64, 0, stream>>>(stats, bn_g + (size_t)i * D,
                                            bn_b + (size_t)i * D, stats + 128,
                                            1.0f / (float)N);
        k_bn_apply_pool<<<cdiv((long long)N * D, 256), 256, 0, stream>>>(
            bufZ, stats + 128, bufH, batch, pooled, i, N);
        h_in = bufH;
    }

    // projection head: relu(pooled @ Wp1 + bp1) @ Wp2 + bp2
    k_gemm_wmma<<<gemmGridP, 256, 0, stream>>>(pooled, Wp1, bp1, nullptr,
                                               tproj, NG, LD, LD, 1);
    k_gemm_wmma<<<gemmGridP, 256, 0, stream>>>(tproj, Wp2, bp2, nullptr,
                                               out, NG, LD, LD, 0);
}